// StainNormalizationLayer_28338194219467
// MI455X (gfx1250) — compile-verified
//
#include <hip/hip_runtime.h>

// ---------------------------------------------------------------------------
// StainNormalization on MI455X (gfx1250, wave32).
//
// out = clip(exp(-(gamma * ((-log(x+1e-6)) @ M) + beta)), 0, 1)
//     = min(exp2( A @ log2(x+1e-6) - beta/ln2 ), 1)   with A[d][c]=gamma_d*M[c][d]
//
// Memory-bound: 402 MB traffic -> ~17.3 us floor at 23.3 TB/s. The 3x3 matvec
// is one V_WMMA_F32_16X16X4_F32 per 16 pixels; B is assembled in WMMA VGPR
// layout straight from interleaved-RGB b96 loads (no permutes) and D stores
// straight back to interleaved RGB (lane p -> 3 contiguous floats -> b96).
// Fast path: wave-uniform, 32-bit indexing, 8 loads issued before first use
// (deep MLP); clamped 64-bit tail path only for the final partial wave.
// ---------------------------------------------------------------------------

typedef __attribute__((ext_vector_type(2)))  float v2f;
typedef __attribute__((ext_vector_type(8)))  float v8f;

struct F3 { float x, y, z; };

#define TILES 8                      // 16 pixels per WMMA tile -> 128 px/wave

__global__ __launch_bounds__(256) void stain_norm_wmma_kernel(
    const float* __restrict__ x,      // (npix, 3) interleaved RGB
    const float* __restrict__ sm,     // 3x3 row-major: sm[c*3+d]
    const float* __restrict__ gamma,  // (3,)
    const float* __restrict__ beta,   // (3,)
    float* __restrict__ out,          // (npix, 3)
    int npix)
{
    const int tid  = blockIdx.x * blockDim.x + threadIdx.x;
    const int lane = threadIdx.x & 31;
    const int wave = tid >> 5;
    const int m    = lane & 15;   // row (A) / pixel-in-tile (B, D)
    const int hi   = lane >> 4;   // half-wave selector (K 0/1 vs K 2/3)

    const float INV_LN2 = 1.4426950408889634f;
    const float bias0 = beta[0] * INV_LN2;
    const float bias1 = beta[1] * INV_LN2;
    const float bias2 = beta[2] * INV_LN2;

    // ---- A matrix (16x4 f32) in WMMA layout: A[d][c] = gamma[d] * sm[c*3+d]
    // VGPR0: lanes0-15 -> K=0, lanes16-31 -> K=2
    // VGPR1: lanes0-15 -> K=1, lanes16-31 -> K=3 (zero pad row)
    float a0 = 0.0f, a1 = 0.0f;
    if (m < 3) {
        const float g = gamma[m];
        a0 = g * sm[(2 * hi) * 3 + m];        // c = 0 (lo) or c = 2 (hi)
        if (hi == 0) a1 = g * sm[3 + m];      // c = 1 ; c = 3 stays 0
    }
    v2f A; A.x = a0; A.y = a1;

    const int base = wave * (TILES * 16);     // first pixel of this wave

    if (base + TILES * 16 <= npix) {
        // ------------------- fast path (wave-uniform, 32-bit indexing) -----
        // Stage all tile loads first: up to 8 global_load_b96 in flight.
        F3 v[TILES];
        #pragma unroll
        for (int u = 0; u < TILES; ++u) {
            const int pm = base + u * 16 + m;         // this lane's pixel
            v[u] = *(const F3*)(x + (size_t)((unsigned)pm * 3u));
        }

        #pragma unroll
        for (int u = 0; u < TILES; ++u) {
            // B (4x16 f32) directly in WMMA layout:
            //   V0: lanes0-15 = log2(r), lanes16-31 = log2(b)   (K=0 / K=2)
            //   V1: lanes0-15 = log2(g), lanes16-31 = finite pad (K=3 -> A=0)
            const float e0 = hi ? v[u].z : v[u].x;
            const float e1 = hi ? v[u].z : v[u].y;
            v2f B;
            B.x = __builtin_amdgcn_logf(e0 + 1e-6f);  // v_log_f32 (log2)
            B.y = __builtin_amdgcn_logf(e1 + 1e-6f);

            v8f C = {};
            v8f D = __builtin_amdgcn_wmma_f32_16x16x4_f32(
                false, A, false, B, (short)0, C, false, false);

            const float o0 = fminf(__builtin_amdgcn_exp2f(D[0] - bias0), 1.0f);
            const float o1 = fminf(__builtin_amdgcn_exp2f(D[1] - bias1), 1.0f);
            const float o2 = fminf(__builtin_amdgcn_exp2f(D[2] - bias2), 1.0f);

            if (hi == 0) {
                const int pm = base + u * 16 + m;
                float* p = out + (size_t)((unsigned)pm * 3u);  // b96 store
                p[0] = o0;
                p[1] = o1;
                p[2] = o2;
            }
        }
    } else {
        // ------------------- tail path (final partial wave only) -----------
        #pragma unroll
        for (int u = 0; u < TILES; ++u) {
            const long pm = (long)base + u * 16 + m;
            const long pc = (pm < npix) ? pm : (long)(npix - 1);
            const F3 v = *(const F3*)(x + pc * 3);

            const float e0 = hi ? v.z : v.x;
            const float e1 = hi ? v.z : v.y;
            v2f B;
            B.x = __builtin_amdgcn_logf(e0 + 1e-6f);
            B.y = __builtin_amdgcn_logf(e1 + 1e-6f);

            v8f C = {};
            v8f D = __builtin_amdgcn_wmma_f32_16x16x4_f32(
                false, A, false, B, (short)0, C, false, false);

            const float o0 = fminf(__builtin_amdgcn_exp2f(D[0] - bias0), 1.0f);
            const float o1 = fminf(__builtin_amdgcn_exp2f(D[1] - bias1), 1.0f);
            const float o2 = fminf(__builtin_amdgcn_exp2f(D[2] - bias2), 1.0f);

            if (hi == 0 && pm < npix) {
                float* p = out + pm * 3;
                p[0] = o0;
                p[1] = o1;
                p[2] = o2;
            }
        }
    }
}

extern "C" void kernel_launch(void* const* d_in, const int* in_sizes, int n_in,
                              void* d_out, int out_size, void* d_ws, size_t ws_size,
                              hipStream_t stream) {
    const float* x     = (const float*)d_in[0];
    const float* sm    = (const float*)d_in[1];
    const float* gamma = (const float*)d_in[2];
    const float* beta  = (const float*)d_in[3];
    float* out = (float*)d_out;

    const int npix = in_sizes[0] / 3;                 // 64*512*512 = 16,777,216
    const int ppw  = TILES * 16;                      // 128 pixels per wave
    const int waves  = (npix + ppw - 1) / ppw;
    const int blocks = (waves + 7) / 8;               // 8 waves (256 thr)/block

    stain_norm_wmma_kernel<<<blocks, 256, 0, stream>>>(x, sm, gamma, beta, out, npix);
}